// MOERouter_35845797053230
// MI455X (gfx1250) — compile-verified
//
#include <hip/hip_runtime.h>

// ---------------------------------------------------------------------------
// MoE router for MI455X (gfx1250): logits GEMM via V_WMMA_F32_16X16X4_F32 with
// double-buffered async global->LDS staging (ASYNCcnt pipelining), fused
// top-2 softmax epilogue.  T=16384, D=4096, E=64, K=2.
// HBM floor: x 268MB + outputs ~13MB @ 23.3 TB/s ~= 12 us.
// ---------------------------------------------------------------------------

typedef __attribute__((ext_vector_type(2))) float v2f;
typedef __attribute__((ext_vector_type(8))) float v8f;
typedef __attribute__((ext_vector_type(4))) float f32x4;
typedef int v4i_g __attribute__((vector_size(16)));   // builtin pointee type

#define T_TOK 16384
#define DDIM  4096
#define NEXP  64
#define KSEL  2
#define KC    64           // K-chunk per LDS buffer
#define LDA   68           // padded row strides (floats): conflict-free b64 reads
#define LDB   68
#define LDL   65
#define A_FLTS (16 * LDA)      // 1088
#define B_FLTS (NEXP * LDB)    // 4352

__device__ __forceinline__ void async_copy16(const float* g, float* l) {
#if __has_builtin(__builtin_amdgcn_global_load_async_to_lds_b128)
  __builtin_amdgcn_global_load_async_to_lds_b128(
      (v4i_g __attribute__((address_space(1)))*)g,
      (v4i_g __attribute__((address_space(3)))*)l,
      0, 0);
#else
  *(f32x4*)l = *(const f32x4*)g;
#endif
}

#if __has_builtin(__builtin_amdgcn_s_wait_asynccnt)
#define WAIT_ASYNC(n) __builtin_amdgcn_s_wait_asynccnt(n)
#else
#define WAIT_ASYNC(n) asm volatile("s_wait_asynccnt %0" ::"i"(n) : "memory")
#endif

// Issue the 10 async 16B transfers (per wave) staging one K-chunk: A 16xKC, B 64xKC.
__device__ __forceinline__ void stage_tiles(const float* __restrict__ x,
                                            const float* __restrict__ gw,
                                            int m0, int kb, float* Ab, float* Bb,
                                            int tid) {
  const int rr = tid >> 4;          // 0..7
  const int cc = (tid & 15) << 2;   // float4 column
#pragma unroll
  for (int it = 0; it < 2; ++it) {  // A: 16 rows
    const int r = rr + (it << 3);
    async_copy16(&x[(size_t)(m0 + r) * DDIM + kb + cc], &Ab[r * LDA + cc]);
  }
#pragma unroll
  for (int it = 0; it < 8; ++it) {  // B: 64 expert rows
    const int e = rr + (it << 3);
    async_copy16(&gw[(size_t)e * DDIM + kb + cc], &Bb[e * LDB + cc]);
  }
}

__device__ __forceinline__ v8f compute_chunk(const float* Ab, const float* Bb,
                                             int aoff, int boff, v8f acc) {
#pragma unroll
  for (int ks = 0; ks < KC; ks += 4) {
    v2f a = *(const v2f*)&Ab[aoff + ks];
    v2f b = *(const v2f*)&Bb[boff + ks];
    acc = __builtin_amdgcn_wmma_f32_16x16x4_f32(false, a, false, b,
                                                (short)0, acc, false, false);
  }
  return acc;
}

__global__ __launch_bounds__(128) void moe_router_kernel(
    const float* __restrict__ x,     // [T, D]
    const float* __restrict__ gw,    // [E, D]
    const float* __restrict__ gb,    // [E]
    float* __restrict__ out)         // logits | weights | indices | mask
{
  __shared__ float smem[2 * A_FLTS + 2 * B_FLTS];      // 43.5 KB
  float* As0 = smem;
  float* As1 = smem + A_FLTS;
  float* Bs0 = smem + 2 * A_FLTS;
  float* Bs1 = smem + 2 * A_FLTS + B_FLTS;
  // Epilogue scratch aliases the (dead) tile buffers after the main loop:
  float* Ls = smem;                                     // 16*LDL = 1040 floats
  int*   Ms = (int*)(smem + 2 * A_FLTS);                // 2048 ints

  const int tid  = threadIdx.x;
  const int lane = tid & 31;
  const int wv   = tid >> 5;                 // wave id 0..3 -> expert tile
  const int m0   = blockIdx.x * 16;          // token tile base

  const int mrow  = lane & 15;
  const int khalf = (lane >> 4) << 1;        // 0 or 2
  const int aoff  = mrow * LDA + khalf;
  const int boff  = (wv * 16 + mrow) * LDB + khalf;

  v8f acc = {};

  // ---- software-pipelined main loop: load chunk c+1 while computing chunk c
  stage_tiles(x, gw, m0, 0, As0, Bs0, tid);
  for (int kb = 0; kb < DDIM; kb += 2 * KC) {
    stage_tiles(x, gw, m0, kb + KC, As1, Bs1, tid);
    WAIT_ASYNC(10);                      // 10 newer in flight => buffer0 resident
    __syncthreads();
    acc = compute_chunk(As0, Bs0, aoff, boff, acc);
    __syncthreads();

    if (kb + 2 * KC < DDIM) {
      stage_tiles(x, gw, m0, kb + 2 * KC, As0, Bs0, tid);
      WAIT_ASYNC(10);                    // buffer1 resident
    } else {
      WAIT_ASYNC(0);
    }
    __syncthreads();
    acc = compute_chunk(As1, Bs1, aoff, boff, acc);
    __syncthreads();
  }

  // ---- epilogue: bias add, stage logits tile in LDS ----
  const int n  = lane & 15;
  const int e  = wv * 16 + n;                // expert column this lane owns
  const int mh = (lane >> 4) << 3;           // 0 or 8 (C layout M offset)
  const float bias = gb[e];

#pragma unroll
  for (int v = 0; v < 8; ++v) {
    Ls[(mh + v) * LDL + e] = acc[v] + bias;
  }
  for (int i = tid; i < NEXP * KSEL * 16; i += 128) Ms[i] = 0;
  __syncthreads();

  // coalesced logits store: 16 rows x 64 floats as b128 per thread
#pragma unroll
  for (int it = 0; it < 2; ++it) {
    const int i = tid + (it << 7);
    const int r = i >> 4;
    const int c = (i & 15) << 2;
    *(f32x4*)&out[(size_t)(m0 + r) * NEXP + c] = *(const f32x4*)&Ls[r * LDL + c];
  }

  // ---- per-token top-2 / weights / indices (threads 0..15) ----
  if (tid < 16) {
    const int t = m0 + tid;
    float l0 = -3.402823466e38f, l1 = -3.402823466e38f;
    int   e0 = 0, e1 = 0;
    for (int ee = 0; ee < NEXP; ++ee) {
      const float lv = Ls[tid * LDL + ee];
      if (lv > l0)      { l1 = l0; e1 = e0; l0 = lv; e0 = ee; }
      else if (lv > l1) { l1 = lv; e1 = ee; }
    }
    // renormalized top-2 softmax weights: w0 = 1/(1+exp(l1-l0))
    const float r = __expf(l1 - l0);
    const float s = 1.0f / (1.0f + r);
    float* wout = out + (size_t)T_TOK * NEXP;
    int*   iout = (int*)(out + (size_t)T_TOK * NEXP + (size_t)T_TOK * KSEL);
    wout[t * KSEL + 0] = s;
    wout[t * KSEL + 1] = r * s;
    iout[t * KSEL + 0] = e0;
    iout[t * KSEL + 1] = e1;
    Ms[(e0 * KSEL + 0) * 16 + tid] = 1;
    Ms[(e1 * KSEL + 1) * 16 + tid] = 1;
  }
  __syncthreads();

  // ---- stream mask tile: mask[(e*K+k)*T + t], 64B per 16 threads ----
  int* mout = (int*)(out + (size_t)T_TOK * NEXP + 2 * (size_t)T_TOK * KSEL);
#pragma unroll
  for (int it = 0; it < 16; ++it) {
    const int i = tid + (it << 7);
    mout[(size_t)(i >> 4) * T_TOK + m0 + (i & 15)] = Ms[i];
  }
}

extern "C" void kernel_launch(void* const* d_in, const int* in_sizes, int n_in,
                              void* d_out, int out_size, void* d_ws, size_t ws_size,
                              hipStream_t stream) {
  const float* x  = (const float*)d_in[0];
  const float* gw = (const float*)d_in[1];
  const float* gb = (const float*)d_in[2];
  float* out = (float*)d_out;

  dim3 grid(T_TOK / 16);
  dim3 block(128);
  hipLaunchKernelGGL(moe_router_kernel, grid, block, 0, stream, x, gw, gb, out);
}